// RQSplineFlow_56349970924099
// MI455X (gfx1250) — compile-verified
//
#include <hip/hip_runtime.h>
#include <hip/hip_bf16.h>
#include <stdint.h>

// RQ-spline flow, MI455X (gfx1250). Memory-bound: 512 MB traffic -> ~22us floor
// at 23.3 TB/s. No matmul structure -> no WMMA; CDNA5 paths used: async
// global->LDS staging (ASYNCcnt) + 320KB-WGP LDS-resident packed knot tables.

#define DDIM   64
#define NB     32
#define RMIN  -5.0f

// LDS layout (per block):
//  sPRM : staged raw params  (widths 2048 f, heights 2048 f, slopes 1984 f)
//  sA   : (kx[k], kx[k+1], ky[k], ky[k+1])  stride 33 float4 per dim (bank pad)
//  sB   : (kd[k], kd[k+1])                  stride 33 float2 per dim
//  sQ   : group knots (kx[4g+1..4g+4])      stride  9 float4 per dim
//  sKX  : plain kx[0..32]                   stride 33 float  per dim

__device__ __forceinline__ float rq_spline_eval(
    float x,
    const float4* __restrict__ Arow,
    const float2* __restrict__ Brow,
    const float4* __restrict__ Qrow,
    const float* __restrict__ kc,   // 7 coarse knots kx[4],kx[8],...,kx[28] (registers)
    float kxlast)
{
    // level 1: coarse group (monotone count, branchless, register-only)
    int g = 0;
#pragma unroll
    for (int j = 0; j < 7; ++j) g += (x >= kc[j]) ? 1 : 0;
    // level 2: within group of 4 bins
    float4 q = Qrow[g];                 // kx[4g+1], kx[4g+2], kx[4g+3], kx[4g+4]
    int k = 4 * g;
    k += (x >= q.x) ? 1 : 0;
    k += (x >= q.y) ? 1 : 0;
    k += (x >= q.z) ? 1 : 0;           // k in [0,31]

    float4 a = Arow[k];                 // kx_k, kx_k1, ky_k, ky_k1
    float2 b = Brow[k];                 // kd_k, kd_k1

    float w    = a.y - a.x;
    float h    = a.w - a.z;
    float invw = __builtin_amdgcn_rcpf(w);
    float s    = h * invw;
    float t    = (x - a.x) * invw;
    float tt   = t * (1.0f - t);
    float num  = h * (s * t * t + b.x * tt);
    float den  = s + (b.y + b.x - 2.0f * s) * tt;
    float y    = a.z + num * __builtin_amdgcn_rcpf(den);

    bool inside = (x >= RMIN) && (x <= kxlast);
    return inside ? y : x;
}

__global__ __launch_bounds__(256) void rqspline_kernel(
    const float* __restrict__ X,
    const float* __restrict__ BW,   // [64][32]
    const float* __restrict__ BH,   // [64][32]
    const float* __restrict__ KS,   // [64][31]
    float* __restrict__ Y,
    long long n)
{
    __shared__ float  sPRM[2048 + 2048 + 2048];
    __shared__ float4 sA[DDIM * 33];
    __shared__ float2 sB[DDIM * 33];
    __shared__ float4 sQ[DDIM * 9];
    __shared__ float  sKX[DDIM * 33];

    const int tid = threadIdx.x;

    // ---- stage raw params into LDS with gfx1250 async global->LDS b128 ----
    {
        for (int i = tid; i < 512; i += 256) {            // widths: 8192 B
            unsigned            l = (unsigned)(size_t)(&sPRM[0] + i * 4);
            unsigned long long  g = (unsigned long long)(const void*)(BW + i * 4);
            asm volatile("global_load_async_to_lds_b128 %0, %1, off"
                         :: "v"(l), "v"(g) : "memory");
        }
        for (int i = tid; i < 512; i += 256) {            // heights: 8192 B
            unsigned            l = (unsigned)(size_t)(&sPRM[2048] + i * 4);
            unsigned long long  g = (unsigned long long)(const void*)(BH + i * 4);
            asm volatile("global_load_async_to_lds_b128 %0, %1, off"
                         :: "v"(l), "v"(g) : "memory");
        }
        for (int i = tid; i < 496; i += 256) {            // slopes: 7936 B
            unsigned            l = (unsigned)(size_t)(&sPRM[4096] + i * 4);
            unsigned long long  g = (unsigned long long)(const void*)(KS + i * 4);
            asm volatile("global_load_async_to_lds_b128 %0, %1, off"
                         :: "v"(l), "v"(g) : "memory");
        }
        asm volatile("s_wait_asynccnt 0" ::: "memory");
    }
    __syncthreads();

    // ---- build packed knot tables (thread d handles dimension d) ----
    if (tid < DDIM) {
        const int d = tid;
        const float* W = &sPRM[0    + d * 32];
        const float* H = &sPRM[2048 + d * 32];
        const float* S = &sPRM[4096 + d * 31];
        float cx = RMIN, cy = RMIN;
        sKX[d * 33 + 0] = cx;
#pragma unroll 1
        for (int i = 0; i < NB; ++i) {
            float nx = cx + W[i];
            float ny = cy + H[i];
            float d0 = (i == 0)      ? 1.0f : S[i - 1];
            float d1 = (i == NB - 1) ? 1.0f : S[i];
            sA[d * 33 + i] = make_float4(cx, nx, cy, ny);
            sB[d * 33 + i] = make_float2(d0, d1);
            sKX[d * 33 + i + 1] = nx;
            cx = nx; cy = ny;
        }
#pragma unroll
        for (int g = 0; g < 8; ++g) {
            sQ[d * 9 + g] = make_float4(sKX[d * 33 + 4 * g + 1],
                                        sKX[d * 33 + 4 * g + 2],
                                        sKX[d * 33 + 4 * g + 3],
                                        sKX[d * 33 + 4 * g + 4]);
        }
    }
    __syncthreads();

    // ---- per-thread constants: fixed dimension, coarse knots in registers ----
    const int d = tid & 63;   // gridDim*blockDim is a multiple of 64
    float kc[7];
#pragma unroll
    for (int j = 0; j < 7; ++j) kc[j] = sKX[d * 33 + 4 * (j + 1)];
    const float  kxlast = sKX[d * 33 + 32];
    const float4* __restrict__ Arow = &sA[d * 33];
    const float2* __restrict__ Brow = &sB[d * 33];
    const float4* __restrict__ Qrow = &sQ[d * 9];

    const long long stride = (long long)gridDim.x * (long long)blockDim.x;
    long long e = (long long)blockIdx.x * blockDim.x + tid;

    // 4-way unrolled grid-stride loop (all 4 elements share this thread's d)
    for (; e + 3 * stride < n; e += 4 * stride) {
        float x0 = X[e];
        float x1 = X[e + stride];
        float x2 = X[e + 2 * stride];
        float x3 = X[e + 3 * stride];
        float y0 = rq_spline_eval(x0, Arow, Brow, Qrow, kc, kxlast);
        float y1 = rq_spline_eval(x1, Arow, Brow, Qrow, kc, kxlast);
        float y2 = rq_spline_eval(x2, Arow, Brow, Qrow, kc, kxlast);
        float y3 = rq_spline_eval(x3, Arow, Brow, Qrow, kc, kxlast);
        Y[e]              = y0;
        Y[e + stride]     = y1;
        Y[e + 2 * stride] = y2;
        Y[e + 3 * stride] = y3;
    }
    for (; e < n; e += stride) {
        Y[e] = rq_spline_eval(X[e], Arow, Brow, Qrow, kc, kxlast);
    }
}

extern "C" void kernel_launch(void* const* d_in, const int* in_sizes, int n_in,
                              void* d_out, int out_size, void* d_ws, size_t ws_size,
                              hipStream_t stream) {
    const float* x  = (const float*)d_in[0];
    const float* bw = (const float*)d_in[1];
    const float* bh = (const float*)d_in[2];
    const float* ks = (const float*)d_in[3];
    float* y = (float*)d_out;
    long long n = (long long)in_sizes[0];   // N*D = 1<<26

    // 2048 blocks * 256 threads = 524288 threads (multiple of 64 -> fixed d/thread)
    rqspline_kernel<<<dim3(2048), dim3(256), 0, stream>>>(x, bw, bh, ks, y, n);
}